// HungarianMatcher_72653666779388
// MI455X (gfx1250) — compile-verified
//
#include <hip/hip_runtime.h>

#define B_    256
#define Q_    900
#define C_    80
#define T_    300
#define QTILE 64
#define NQT   15           // ceil(900/64)
#define NTHREADS 256

__global__ __launch_bounds__(NTHREADS) void
matcher_cost_kernel(const float* __restrict__ logits,   // [B,Q,C]
                    const float* __restrict__ pboxes,   // [B,Q,4] cxcywh
                    const int*   __restrict__ labels,   // [B,T]
                    const float* __restrict__ tboxes,   // [B,T,4] cxcywh
                    float* __restrict__ out)            // [B,Q,T]
{
    // LDS staging (~33.5 KB total)
    __shared__ __align__(16) float prob[QTILE * C_];   // 20480 B: logits -> (-2*sigmoid) in place
    __shared__ float4 t_cxcywh[T_];                    // 4800 B
    __shared__ float4 t_xyxy[T_];                      // 4800 B
    __shared__ float  t_area[T_];                      // 1200 B
    __shared__ int    t_lab[T_];                       // 1200 B
    __shared__ float4 p_box[QTILE];                    // 1024 B

    const int b   = blockIdx.x / NQT;
    const int qt  = blockIdx.x % NQT;
    const int q0  = qt * QTILE;
    const int tid = threadIdx.x;

    int rows = Q_ - q0; if (rows > QTILE) rows = QTILE;   // last tile has 4 rows
    const int tile_bytes = rows * C_ * 4;                 // multiple of 16 (C_*4 = 320)

    // ---------------- Phase 1a: async DMA logits tile -> LDS (ASYNCcnt path) ----
    {
        const float* gsrc = logits + ((size_t)b * Q_ + q0) * C_;   // 16B-aligned
        // generic shared address: low 32 bits are the LDS byte offset (ISA 10.2)
        unsigned lds_base = (unsigned)(size_t)(void*)&prob[0];
        #pragma unroll 1
        for (int off = tid * 16; off < tile_bytes; off += NTHREADS * 16) {
            unsigned ldsa = lds_base + (unsigned)off;
            unsigned goff = (unsigned)off;
            asm volatile("global_load_async_to_lds_b128 %0, %1, %2"
                         :
                         : "v"(ldsa), "v"(goff), "s"(gsrc)
                         : "memory");
        }
    }

    // ---------------- Phase 1b: stage target + pred boxes (overlaps async DMA) --
    for (int t = tid; t < T_; t += NTHREADS) {
        float4 tb = ((const float4*)tboxes)[(size_t)b * T_ + t];
        float hw = 0.5f * tb.z, hh = 0.5f * tb.w;
        t_cxcywh[t] = tb;
        t_xyxy[t]   = make_float4(tb.x - hw, tb.y - hh, tb.x + hw, tb.y + hh);
        t_area[t]   = tb.z * tb.w;
        t_lab[t]    = labels[(size_t)b * T_ + t];
    }
    if (tid < rows) {
        p_box[tid] = ((const float4*)pboxes)[(size_t)b * Q_ + q0 + tid];
    }

    // wait for async tile, then transform logits -> -2*sigmoid in place
    asm volatile("s_wait_asynccnt 0" ::: "memory");
    __syncthreads();

    {
        const int n = rows * C_;
        for (int i = tid; i < n; i += NTHREADS) {
            float x = prob[i];
            prob[i] = -2.0f * __builtin_amdgcn_rcpf(1.0f + __expf(-x));
        }
    }
    __syncthreads();

    // ---------------- Phase 2: each wave sweeps queries; lanes own 4 targets ----
    const int wave = tid >> 5;
    const int lane = tid & 31;

    for (int ql = wave; ql < QTILE; ql += 8) {
        const int q = q0 + ql;
        if (q >= Q_) break;

        float4 pb = p_box[ql];                              // ds broadcast
        const float phw = 0.5f * pb.z, phh = 0.5f * pb.w;
        const float px0 = pb.x - phw, py0 = pb.y - phh;
        const float px1 = pb.x + phw, py1 = pb.y + phh;
        const float parea = pb.z * pb.w;

        const float* probq = &prob[ql * C_];
        float* __restrict__ orow = out + ((size_t)b * Q_ + q) * T_;

        #pragma unroll 1
        for (int c = 0; c < 3; ++c) {
            const int t0 = c * 128 + lane * 4;
            if (t0 < T_) {                                  // chunk 2: lanes 0..10 (44 = 11*4)
                float r[4];
                #pragma unroll
                for (int k = 0; k < 4; ++k) {
                    const int t = t0 + k;
                    float4 tb = t_cxcywh[t];                // ds_load_b128
                    float4 tx = t_xyxy[t];                  // ds_load_b128
                    float  ta = t_area[t];
                    float  pc2 = probq[t_lab[t]];           // LDS gather, already -2*sigmoid

                    // L1 in cxcywh space
                    float cb = fabsf(pb.x - tb.x) + fabsf(pb.y - tb.y)
                             + fabsf(pb.z - tb.z) + fabsf(pb.w - tb.w);

                    // IoU
                    float ltx = fmaxf(px0, tx.x), lty = fmaxf(py0, tx.y);
                    float rbx = fminf(px1, tx.z), rby = fminf(py1, tx.w);
                    float iw  = fmaxf(rbx - ltx, 0.0f);
                    float ih  = fmaxf(rby - lty, 0.0f);
                    float inter = iw * ih;
                    float uni   = parea + ta - inter;
                    float iou   = inter * __builtin_amdgcn_rcpf(uni);

                    // enclosing box
                    float cx0 = fminf(px0, tx.x), cy0 = fminf(py0, tx.y);
                    float cx1 = fmaxf(px1, tx.z), cy1 = fmaxf(py1, tx.w);
                    float cw  = fmaxf(cx1 - cx0, 0.0f);
                    float chh = fmaxf(cy1 - cy0, 0.0f);
                    float ac  = cw * chh;
                    float giou = iou - (ac - uni) * __builtin_amdgcn_rcpf(ac);

                    // 2*(-sigmoid) + 5*L1 + 2*(-giou)
                    r[k] = fmaf(5.0f, cb, fmaf(-2.0f, giou, pc2));
                }
                *(float4*)(orow + t0) = make_float4(r[0], r[1], r[2], r[3]);  // b128 store
            }
        }
    }
}

extern "C" void kernel_launch(void* const* d_in, const int* in_sizes, int n_in,
                              void* d_out, int out_size, void* d_ws, size_t ws_size,
                              hipStream_t stream) {
    const float* logits = (const float*)d_in[0];   // [B,Q,C] f32
    const float* pboxes = (const float*)d_in[1];   // [B,Q,4] f32
    const int*   labels = (const int*)d_in[2];     // [B,T] i32
    const float* tboxes = (const float*)d_in[3];   // [B,T,4] f32
    float* out = (float*)d_out;

    const int B = in_sizes[0] / (Q_ * C_);         // 256
    dim3 grid(B * NQT);
    dim3 block(NTHREADS);
    matcher_cost_kernel<<<grid, block, 0, stream>>>(logits, pboxes, labels, tboxes, out);
}